// MoleculeEncoder_AttnFP_82016695485246
// MI455X (gfx1250) — compile-verified
//
#include <hip/hip_runtime.h>
#include <hip/hip_bf16.h>
#include <math.h>

#define N_NODES 262144
#define N_EDGES 2097152
#define N_GRAPHS 8192
#define IN_CH 54
#define HID 32
#define OUT_CH 64
#define SLOPE 0.01f

typedef float v2f __attribute__((ext_vector_type(2)));
typedef float v8f __attribute__((ext_vector_type(8)));

__device__ __forceinline__ float leakyf(float x) { return x > 0.f ? x : SLOPE * x; }
__device__ __forceinline__ float eluf(float x)   { return x > 0.f ? x : expm1f(x); }
__device__ __forceinline__ float mishf(float x) {
    float sp = (x > 20.f) ? x : log1pf(__expf(x));
    return x * tanhf(sp);
}
__device__ __forceinline__ float sigm(float x) { return 1.f / (1.f + __expf(-x)); }

// float atomic max via ordered int/uint trick (works for mixed signs, init = -inf)
__device__ __forceinline__ void atomicMaxF(float* a, float v) {
    if (v >= 0.f) atomicMax((int*)a, __float_as_int(v));
    else          atomicMin((unsigned int*)a, __float_as_uint(v));
}

// ---------------------------------------------------------------- utilities
__global__ __launch_bounds__(256) void k_fill(float* p, long n, float v) {
    long i = (long)blockIdx.x * blockDim.x + threadIdx.x;
    if (i < n) p[i] = v;
}
__global__ __launch_bounds__(256) void k_relu(float* p, long n) {
    long i = (long)blockIdx.x * blockDim.x + threadIdx.x;
    if (i < n) p[i] = fmaxf(p[i], 0.f);
}

// U[r][c] = act( U[r][c] / (s[r]+1e-16) + bias[c] )   (bias may be null; act 0=none, 2=elu)
__global__ __launch_bounds__(256) void k_div_act(float* U, const float* __restrict__ s,
                                                 const float* __restrict__ bias,
                                                 long rows, int act) {
    long i = (long)blockIdx.x * blockDim.x + threadIdx.x;
    if (i >= rows * HID) return;
    long r = i >> 5; int c = (int)(i & 31);
    float v = U[i] / (s[r] + 1e-16f);
    if (bias) v += bias[c];
    if (act == 2) v = eluf(v);
    U[i] = v;
}

// ---------------------------------------------------------------- WMMA GEMM
// C[M x (32*colTiles)] = act( A[M x K] @ W[N x K]^T + bias ), f32 via v_wmma_f32_16x16x4_f32
// Each wave: 16 rows x 32 cols.  A operand layout (ISA 7.12.2): VGPR v = K=v (lanes 0-15,
// M=lane) / K=v+2 (lanes 16-31, M=lane-16); B symmetric with N=lane&15.
// K is a template parameter: the K%4==0 cases unroll into a branch-free load+wmma stream;
// the K=54 tail uses clamped addresses + cndmask selects (no EXEC-masked loads).
template <int K>
__global__ __launch_bounds__(256) void k_gemm_wmma(const float* __restrict__ A, int lda,
                                                   const float* __restrict__ W, int ldw,
                                                   const float* __restrict__ bias,
                                                   float* __restrict__ C, int ldc,
                                                   int rowTiles, int colTiles, int act) {
    int wid = (blockIdx.x * blockDim.x + threadIdx.x) >> 5;
    if (wid >= rowTiles * colTiles) return;
    int lane = threadIdx.x & 31;
    int rt = wid % rowTiles, ct = wid / rowTiles;
    int row0 = rt * 16, c0 = ct * 32;
    int m = lane & 15, half = lane >> 4;

    const float* arow  = A + (size_t)(row0 + m) * lda;
    const float* wrow0 = W + (size_t)(c0 + m) * ldw;
    const float* wrow1 = W + (size_t)(c0 + 16 + m) * ldw;

    v8f acc0 = {}; v8f acc1 = {};
    constexpr int KFULL = K & ~3;
#pragma unroll
    for (int k0 = 0; k0 < KFULL; k0 += 4) {
        int ka = k0 + half * 2;
        v2f a, b0, b1;
        a.x  = arow[ka];   a.y  = arow[ka + 1];
        b0.x = wrow0[ka];  b0.y = wrow0[ka + 1];
        b1.x = wrow1[ka];  b1.y = wrow1[ka + 1];
        acc0 = __builtin_amdgcn_wmma_f32_16x16x4_f32(false, a, false, b0, 0, acc0, false, false);
        acc1 = __builtin_amdgcn_wmma_f32_16x16x4_f32(false, a, false, b1, 0, acc1, false, false);
    }
    if constexpr ((K & 3) != 0) {
        // tail: clamp addresses in-bounds, select 0 for k >= K (pure VALU cndmask, no branches)
        int ka = KFULL + half * 2;
        bool p0 = (ka < K), p1 = (ka + 1 < K);
        int kc0 = p0 ? ka : 0;
        int kc1 = p1 ? (ka + 1) : 0;
        v2f a, b0, b1;
        a.x  = p0 ? arow[kc0]  : 0.f;  a.y  = p1 ? arow[kc1]  : 0.f;
        b0.x = p0 ? wrow0[kc0] : 0.f;  b0.y = p1 ? wrow0[kc1] : 0.f;
        b1.x = p0 ? wrow1[kc0] : 0.f;  b1.y = p1 ? wrow1[kc1] : 0.f;
        acc0 = __builtin_amdgcn_wmma_f32_16x16x4_f32(false, a, false, b0, 0, acc0, false, false);
        acc1 = __builtin_amdgcn_wmma_f32_16x16x4_f32(false, a, false, b1, 0, acc1, false, false);
    }
    int col0 = c0 + m, col1 = c0 + 16 + m;
    float bb0 = bias ? bias[col0] : 0.f;
    float bb1 = bias ? bias[col1] : 0.f;
#pragma unroll
    for (int v = 0; v < 8; ++v) {
        int row = row0 + v + half * 8;
        float x0 = acc0[v] + bb0, x1 = acc1[v] + bb1;
        if (act == 1)      { x0 = leakyf(x0); x1 = leakyf(x1); }
        else if (act == 2) { x0 = eluf(x0);   x1 = eluf(x1);   }
        else if (act == 3) { x0 = mishf(x0);  x1 = mishf(x1);  }
        C[(size_t)row * ldc + col0] = x0;
        C[(size_t)row * ldc + col1] = x1;
    }
}

// ---------------------------------------------------------------- per-row dot products
// o1[r] = F[r]·v1, o2[r] = F[r]·v2 (v2/o2 optional). Wave per row, lane = channel.
__global__ __launch_bounds__(256) void k_dot2(const float* __restrict__ F,
                                              const float* __restrict__ v1,
                                              const float* __restrict__ v2,
                                              float* __restrict__ o1, float* __restrict__ o2,
                                              long rows) {
    long wid = ((long)blockIdx.x * blockDim.x + threadIdx.x) >> 5;
    int lane = threadIdx.x & 31;
    if (wid >= rows) return;
    float f = F[wid * HID + lane];
    float p1 = f * v1[lane];
    float p2 = v2 ? f * v2[lane] : 0.f;
    for (int off = 16; off >= 1; off >>= 1) {
        p1 += __shfl_xor(p1, off, 32);
        p2 += __shfl_xor(p2, off, 32);
    }
    if (lane == 0) { o1[wid] = p1; if (o2) o2[wid] = p2; }
}

// ---------------------------------------------------------------- GATEConv edge pass A
// logit[e] = leaky( sum_c leaky(t[src][c] + ea*W1[c][32]) * att_l[c] + r[dst] )
__global__ __launch_bounds__(256) void k_gate_edgeA(const float* __restrict__ T,
                                                    const float* __restrict__ edge_attr,
                                                    const float* __restrict__ g_lin1_w,
                                                    const float* __restrict__ att_l,
                                                    const float* __restrict__ r,
                                                    const int* __restrict__ src,
                                                    const int* __restrict__ dst,
                                                    float* __restrict__ elog,
                                                    float* __restrict__ nmax, long nE) {
    long e = ((long)blockIdx.x * blockDim.x + threadIdx.x) >> 5;
    int lane = threadIdx.x & 31;
    if (e >= nE) return;
    int s = src[e], d = dst[e];
    float ea = edge_attr[e];
    float ev = leakyf(T[(size_t)s * HID + lane] + ea * g_lin1_w[lane * (HID + 1) + HID]);
    float p = ev * att_l[lane];
    for (int off = 16; off >= 1; off >>= 1) p += __shfl_xor(p, off, 32);
    if (lane == 0) {
        float lg = leakyf(p + r[d]);
        elog[e] = lg;
        atomicMaxF(&nmax[d], lg);
    }
}

// ---------------------------------------------------------------- GATConv edge pass A
__global__ __launch_bounds__(256) void k_gat_edgeA(const float* __restrict__ al,
                                                   const float* __restrict__ ar,
                                                   const int* __restrict__ src,
                                                   const int* __restrict__ dst,
                                                   float* __restrict__ elog,
                                                   float* __restrict__ nmax, long nE) {
    long e = (long)blockIdx.x * blockDim.x + threadIdx.x;
    if (e >= nE) return;
    int d = dst[e];
    float lg = leakyf(al[src[e]] + ar[d]);
    elog[e] = lg;
    atomicMaxF(&nmax[d], lg);
}

// ---------------------------------------------------------------- edge pass B (scatter)
// w = exp(logit - max[dst]); nsum[dst]+=w; U[dst] += feat[src]*w.  Wave per edge.
__global__ __launch_bounds__(256) void k_edge_scatter(const float* __restrict__ feat,
                                                      const float* __restrict__ elog,
                                                      const float* __restrict__ nmax,
                                                      const int* __restrict__ src,
                                                      const int* __restrict__ dst,
                                                      float* __restrict__ nsum,
                                                      float* __restrict__ U, long nE) {
    long e = ((long)blockIdx.x * blockDim.x + threadIdx.x) >> 5;
    int lane = threadIdx.x & 31;
    if (e >= nE) return;
    int s = src[e], d = dst[e];
    float w = __expf(elog[e] - nmax[d]);
    if (lane == 0) atomicAdd(&nsum[d], w);
    atomicAdd(&U[(size_t)d * HID + lane], feat[(size_t)s * HID + lane] * w);
}

// ---------------------------------------------------------------- graph readout scatters
__global__ __launch_bounds__(256) void k_seg_sum(const float* __restrict__ X,
                                                 const int* __restrict__ batch,
                                                 float* __restrict__ OUTg, long rows) {
    long n = ((long)blockIdx.x * blockDim.x + threadIdx.x) >> 5;
    int lane = threadIdx.x & 31;
    if (n >= rows) return;
    int g = batch[n];
    atomicAdd(&OUTg[(size_t)g * HID + lane], X[n * HID + lane]);
}

__global__ __launch_bounds__(256) void k_readout_logits(const float* __restrict__ a_node,
                                                        const float* __restrict__ a_g,
                                                        const int* __restrict__ batch,
                                                        float* __restrict__ nlog,
                                                        float* __restrict__ gmax, long rows) {
    long n = (long)blockIdx.x * blockDim.x + threadIdx.x;
    if (n >= rows) return;
    int g = batch[n];
    float lg = leakyf(a_node[n] + a_g[g]);
    nlog[n] = lg;
    atomicMaxF(&gmax[g], lg);
}

__global__ __launch_bounds__(256) void k_readout_scatter(const float* __restrict__ xs,
                                                         const float* __restrict__ nlog,
                                                         const float* __restrict__ gmax,
                                                         const int* __restrict__ batch,
                                                         float* __restrict__ gsum,
                                                         float* __restrict__ gmsg, long rows) {
    long n = ((long)blockIdx.x * blockDim.x + threadIdx.x) >> 5;
    int lane = threadIdx.x & 31;
    if (n >= rows) return;
    int g = batch[n];
    float w = __expf(nlog[n] - gmax[g]);
    if (lane == 0) atomicAdd(&gsum[g], w);
    atomicAdd(&gmsg[(size_t)g * HID + lane], xs[n * HID + lane] * w);
}

// ---------------------------------------------------------------- fused GRU cell + relu
// out = relu( (1-z)*n + z*h ),  gates per torch.nn.GRUCell. Wave per row, lane = channel.
// Weights staged in LDS transposed ([k][96]) for conflict-free reads.
__global__ __launch_bounds__(256) void k_gru(const float* __restrict__ xin,
                                             const float* __restrict__ hid,
                                             float* __restrict__ outp,
                                             const float* __restrict__ wih,
                                             const float* __restrict__ whh,
                                             const float* __restrict__ bih,
                                             const float* __restrict__ bhh, long rows) {
    __shared__ float sWih[HID * 3 * HID];   // [k][96]
    __shared__ float sWhh[HID * 3 * HID];
    for (int i = threadIdx.x; i < 3 * HID * HID; i += blockDim.x) {
        int j = i / HID, k = i % HID;       // wih row-major [96][32]
        sWih[k * (3 * HID) + j] = wih[i];
        sWhh[k * (3 * HID) + j] = whh[i];
    }
    __syncthreads();
    int lane = threadIdx.x & 31;
    long wid = ((long)blockIdx.x * blockDim.x + threadIdx.x) >> 5;
    long nw  = ((long)gridDim.x * blockDim.x) >> 5;
    float bi0 = bih[lane], bi1 = bih[HID + lane], bi2 = bih[2 * HID + lane];
    float bh0 = bhh[lane], bh1 = bhh[HID + lane], bh2 = bhh[2 * HID + lane];
    for (long row = wid; row < rows; row += nw) {
        float xv = xin[row * HID + lane];
        float hv = hid[row * HID + lane];
        float ir = bi0, iz = bi1, inn = bi2;
        float hr = bh0, hz = bh1, hn  = bh2;
#pragma unroll 8
        for (int k = 0; k < HID; ++k) {
            float xk = __shfl(xv, k, 32);
            float hk = __shfl(hv, k, 32);
            const float* wi = &sWih[k * (3 * HID)];
            const float* wh = &sWhh[k * (3 * HID)];
            ir += xk * wi[lane]; iz += xk * wi[HID + lane]; inn += xk * wi[2 * HID + lane];
            hr += hk * wh[lane]; hz += hk * wh[HID + lane]; hn  += hk * wh[2 * HID + lane];
        }
        float rg = sigm(ir + hr);
        float zg = sigm(iz + hz);
        float ng = tanhf(inn + rg * hn);
        float o  = (1.f - zg) * ng + zg * hv;
        outp[row * HID + lane] = fmaxf(o, 0.f);
    }
}

// ================================================================ host driver
extern "C" void kernel_launch(void* const* d_in, const int* in_sizes, int n_in,
                              void* d_out, int out_size, void* d_ws, size_t ws_size,
                              hipStream_t stream) {
    const float* x_in      = (const float*)d_in[0];
    const float* edge_attr = (const float*)d_in[1];
    const int*   eidx      = (const int*)d_in[2];
    const int*   batch     = (const int*)d_in[3];
    const float* lin1_w    = (const float*)d_in[4];
    const float* lin1_b    = (const float*)d_in[5];
    const float* g_lin1_w  = (const float*)d_in[6];
    const float* g_lin2_w  = (const float*)d_in[7];
    const float* g_att_l   = (const float*)d_in[8];
    const float* g_att_r   = (const float*)d_in[9];
    const float* g_bias    = (const float*)d_in[10];
    const float* gru1_wih  = (const float*)d_in[11];
    const float* gru1_whh  = (const float*)d_in[12];
    const float* gru1_bih  = (const float*)d_in[13];
    const float* gru1_bhh  = (const float*)d_in[14];
    const float* a_lin_w   = (const float*)d_in[15];
    const float* a_att_src = (const float*)d_in[16];
    const float* a_att_dst = (const float*)d_in[17];
    const float* a_bias    = (const float*)d_in[18];
    const float* gru2_wih  = (const float*)d_in[19];
    const float* gru2_whh  = (const float*)d_in[20];
    const float* gru2_bih  = (const float*)d_in[21];
    const float* gru2_bhh  = (const float*)d_in[22];
    const float* m_lin_w   = (const float*)d_in[23];
    const float* m_att_src = (const float*)d_in[24];
    const float* m_att_dst = (const float*)d_in[25];
    const float* m_bias    = (const float*)d_in[26];
    const float* gru3_wih  = (const float*)d_in[27];
    const float* gru3_whh  = (const float*)d_in[28];
    const float* gru3_bih  = (const float*)d_in[29];
    const float* gru3_bhh  = (const float*)d_in[30];
    const float* lin2_w    = (const float*)d_in[31];
    const float* lin2_b    = (const float*)d_in[32];
    (void)in_sizes; (void)n_in; (void)out_size; (void)ws_size;

    const int* src = eidx;             // edge_index[0] (j)
    const int* dst = eidx + N_EDGES;   // edge_index[1] (i)

    // workspace carve-up (floats, 256B aligned)
    float* ws = (float*)d_ws;
    size_t off = 0;
    auto alloc = [&](size_t nf) { float* p = ws + off; off += (nf + 63) & ~(size_t)63; return p; };
    float* X    = alloc((size_t)N_NODES * HID);   // node features (updated in place)
    float* T    = alloc((size_t)N_NODES * HID);   // t / h1 / xt / xs
    float* U    = alloc((size_t)N_NODES * HID);   // message accumulator
    float* sA   = alloc(N_NODES);                 // r / al / a_node
    float* sB   = alloc(N_NODES);                 // ar / per-node readout logits
    float* nmax = alloc(N_NODES);
    float* nsum = alloc(N_NODES);
    float* elog = alloc(N_EDGES);
    float* OUTg = alloc((size_t)N_GRAPHS * HID);
    float* GH   = alloc((size_t)N_GRAPHS * HID);
    float* gsc  = alloc(N_GRAPHS);
    float* gmax = alloc(N_GRAPHS);
    float* gsum = alloc(N_GRAPHS);
    float* gmsg = alloc((size_t)N_GRAPHS * HID);

    dim3 B(256);
    auto g1 = [](long n)     { return dim3((unsigned)((n + 255) / 256)); };
    auto gw = [](long waves) { return dim3((unsigned)((waves * 32 + 255) / 256)); };
    const int rtN = N_NODES / 16, rtG = N_GRAPHS / 16;

    // ---- atom embedding: X = leaky(x @ lin1_w^T + b)
    k_gemm_wmma<IN_CH><<<gw(rtN), B, 0, stream>>>(x_in, IN_CH, lin1_w, IN_CH, lin1_b, X, HID, rtN, 1, 1);

    // ---- GATEConv ----
    k_fill<<<g1(N_NODES), B, 0, stream>>>(nmax, N_NODES, -INFINITY);
    k_fill<<<g1(N_NODES), B, 0, stream>>>(nsum, N_NODES, 0.f);
    k_fill<<<g1((long)N_NODES * HID), B, 0, stream>>>(U, (long)N_NODES * HID, 0.f);
    // t = X @ W1[:, :32]^T   (ldw = 33)
    k_gemm_wmma<HID><<<gw(rtN), B, 0, stream>>>(X, HID, g_lin1_w, HID + 1, nullptr, T, HID, rtN, 1, 0);
    // r_i = X . g_att_r
    k_dot2<<<gw(N_NODES), B, 0, stream>>>(X, g_att_r, nullptr, sA, nullptr, N_NODES);
    k_gate_edgeA<<<gw(N_EDGES), B, 0, stream>>>(T, edge_attr, g_lin1_w, g_att_l, sA, src, dst, elog, nmax, N_EDGES);
    k_edge_scatter<<<gw(N_EDGES), B, 0, stream>>>(X, elog, nmax, src, dst, nsum, U, N_EDGES);
    k_div_act<<<g1((long)N_NODES * HID), B, 0, stream>>>(U, nsum, nullptr, N_NODES, 0);
    // h = elu( U @ W2^T + g_bias )
    k_gemm_wmma<HID><<<gw(rtN), B, 0, stream>>>(U, HID, g_lin2_w, HID, g_bias, T, HID, rtN, 1, 2);
    k_gru<<<dim3(1024), B, 0, stream>>>(T, X, X, gru1_wih, gru1_whh, gru1_bih, gru1_bhh, N_NODES);

    // ---- GATConv ----
    k_fill<<<g1(N_NODES), B, 0, stream>>>(nmax, N_NODES, -INFINITY);
    k_fill<<<g1(N_NODES), B, 0, stream>>>(nsum, N_NODES, 0.f);
    k_fill<<<g1((long)N_NODES * HID), B, 0, stream>>>(U, (long)N_NODES * HID, 0.f);
    k_gemm_wmma<HID><<<gw(rtN), B, 0, stream>>>(X, HID, a_lin_w, HID, nullptr, T, HID, rtN, 1, 0);
    k_dot2<<<gw(N_NODES), B, 0, stream>>>(T, a_att_src, a_att_dst, sA, sB, N_NODES);
    k_gat_edgeA<<<g1(N_EDGES), B, 0, stream>>>(sA, sB, src, dst, elog, nmax, N_EDGES);
    k_edge_scatter<<<gw(N_EDGES), B, 0, stream>>>(T, elog, nmax, src, dst, nsum, U, N_EDGES);
    k_div_act<<<g1((long)N_NODES * HID), B, 0, stream>>>(U, nsum, a_bias, N_NODES, 2);
    k_gru<<<dim3(1024), B, 0, stream>>>(U, X, X, gru2_wih, gru2_whh, gru2_bih, gru2_bhh, N_NODES);

    // ---- molecule readout ----
    k_fill<<<g1((long)N_GRAPHS * HID), B, 0, stream>>>(OUTg, (long)N_GRAPHS * HID, 0.f);
    k_seg_sum<<<gw(N_NODES), B, 0, stream>>>(X, batch, OUTg, N_NODES);
    k_relu<<<g1((long)N_GRAPHS * HID), B, 0, stream>>>(OUTg, (long)N_GRAPHS * HID);
    // xs = X @ m_lin_w^T ; a_node = xs . m_att_src
    k_gemm_wmma<HID><<<gw(rtN), B, 0, stream>>>(X, HID, m_lin_w, HID, nullptr, T, HID, rtN, 1, 0);
    k_dot2<<<gw(N_NODES), B, 0, stream>>>(T, m_att_src, nullptr, sA, nullptr, N_NODES);

    for (int t = 0; t < 2; ++t) {
        k_gemm_wmma<HID><<<gw(rtG), B, 0, stream>>>(OUTg, HID, m_lin_w, HID, nullptr, GH, HID, rtG, 1, 0);
        k_dot2<<<gw(N_GRAPHS), B, 0, stream>>>(GH, m_att_dst, nullptr, gsc, nullptr, N_GRAPHS);
        k_fill<<<g1(N_GRAPHS), B, 0, stream>>>(gmax, N_GRAPHS, -INFINITY);
        k_fill<<<g1(N_GRAPHS), B, 0, stream>>>(gsum, N_GRAPHS, 0.f);
        k_fill<<<g1((long)N_GRAPHS * HID), B, 0, stream>>>(gmsg, (long)N_GRAPHS * HID, 0.f);
        k_readout_logits<<<g1(N_NODES), B, 0, stream>>>(sA, gsc, batch, sB, gmax, N_NODES);
        k_readout_scatter<<<gw(N_NODES), B, 0, stream>>>(T, sB, gmax, batch, gsum, gmsg, N_NODES);
        k_div_act<<<g1((long)N_GRAPHS * HID), B, 0, stream>>>(gmsg, gsum, m_bias, N_GRAPHS, 2);
        k_gru<<<dim3(64), B, 0, stream>>>(gmsg, OUTg, OUTg, gru3_wih, gru3_whh, gru3_bih, gru3_bhh, N_GRAPHS);
    }

    // ---- final linear + Mish -> d_out (8192 x 64 f32)
    k_gemm_wmma<HID><<<gw((long)rtG * 2), B, 0, stream>>>(OUTg, HID, lin2_w, HID, lin2_b,
                                                          (float*)d_out, OUT_CH, rtG, OUT_CH / 32, 2 + 1);
}